// TGS_82660940579130
// MI455X (gfx1250) — compile-verified
//
#include <hip/hip_runtime.h>
#include <math.h>

typedef float v2f __attribute__((ext_vector_type(2)));
typedef float v8f __attribute__((ext_vector_type(8)));

#define TGN_N  30000
#define TGN_K  20
#define TGN_D  128
#define TGN_TD 384              // 3*D
#define WAVES_PER_BLOCK 2
#define ROWS_PER_BLOCK  (WAVES_PER_BLOCK * 16)

static __device__ __forceinline__ v8f splat8(float x) {
  v8f r = {x, x, x, x, x, x, x, x};
  return r;
}

// cos(x) for |x| <= ~2e4 rad: 2-term Cody-Waite reduction (exact hi-part
// subtraction: n <= 2^12 and c1 has an 8-bit mantissa, so n*c1 and x - n*c1
// are exact in f32), then the hardware v_cos_f32 on the reduced argument.
// Total argument error ~4e-7 rad -> fp32-grade accuracy at ~7 ops instead of
// libm cosf's ~30.
static __device__ __forceinline__ float fast_cos(float x) {
  const float inv2pi = 0.15915494309189535f;
  const float c1 = 6.28125f;                   // 2*pi hi (short mantissa)
  const float c2 = 1.9353071795864769e-3f;     // 2*pi - c1
  const float n = rintf(x * inv2pi);
  float r = fmaf(n, -c1, x);                   // exact
  r = fmaf(n, -c2, r);                         // |r| <= pi (+eps)
  return __cosf(r);                            // v_cos_f32 path, in-range
}

__global__ __launch_bounds__(WAVES_PER_BLOCK * 32)
void tgn_graphsum_fused(const float* __restrict__ node_feat,   // [N,D]
                        const float* __restrict__ timestamps,  // [N]
                        const float* __restrict__ edge_feat,   // [E,D]
                        const int*   __restrict__ neighbors,   // [N,K]
                        const int*   __restrict__ edge_idxs,   // [N,K]
                        const float* __restrict__ edge_times,  // [N,K]
                        const float* __restrict__ time_w,      // [D]
                        const float* __restrict__ time_b,      // [D]
                        const float* __restrict__ W1,          // [3D,D]
                        const float* __restrict__ b1,          // [D]
                        const float* __restrict__ W2,          // [3D,D]
                        const float* __restrict__ b2,          // [D]
                        float* __restrict__ out)               // [N,D]
{
  // One private 16x384 f32 tile per wave; no cross-wave sharing -> no barriers.
  __shared__ float lds[WAVES_PER_BLOCK][16 * TGN_TD];

  const int lane = threadIdx.x & 31;
  const int wave = threadIdx.x >> 5;
  float* S = &lds[wave][0];

  const int n0 = blockIdx.x * ROWS_PER_BLOCK + wave * 16;  // first row of this wave's tile
  const int c4 = lane * 4;                                 // 4 feature columns per lane

  const float4 w4 = *(const float4*)(time_w + c4);
  const float4 b4 = *(const float4*)(time_b + c4);

  // ---------------- Phase A: gather + time-encode + reduce over K ----------------
  // S[m][  0..127] = sum_k node_feat[nbr]
  // S[m][128..255] = sum_k cos(dt*w+b)
  // S[m][256..383] = sum_k edge_feat[eidx]
  for (int m = 0; m < 16; ++m) {
    int n = n0 + m;
    if (n >= TGN_N) n = TGN_N - 1;           // clamp (stores are guarded later)
    const float tq = timestamps[n];
    const int*   nb_p = neighbors  + n * TGN_K;
    const int*   ei_p = edge_idxs  + n * TGN_K;
    const float* et_p = edge_times + n * TGN_K;

    float4 s_nf = make_float4(0.f, 0.f, 0.f, 0.f);
    float4 s_te = make_float4(0.f, 0.f, 0.f, 0.f);
    float4 s_ef = make_float4(0.f, 0.f, 0.f, 0.f);
#pragma unroll 10
    for (int k = 0; k < TGN_K; ++k) {
      const int   nb = nb_p[k];
      const int   ei = ei_p[k];
      const float dt = tq - et_p[k];
      const float4 a = *(const float4*)(node_feat + (size_t)nb * TGN_D + c4);
      const float4 e = *(const float4*)(edge_feat + (size_t)ei * TGN_D + c4);
      s_nf.x += a.x; s_nf.y += a.y; s_nf.z += a.z; s_nf.w += a.w;
      s_ef.x += e.x; s_ef.y += e.y; s_ef.z += e.z; s_ef.w += e.w;
      s_te.x += fast_cos(fmaf(dt, w4.x, b4.x));
      s_te.y += fast_cos(fmaf(dt, w4.y, b4.y));
      s_te.z += fast_cos(fmaf(dt, w4.z, b4.z));
      s_te.w += fast_cos(fmaf(dt, w4.w, b4.w));
    }
    *(float4*)&S[m * TGN_TD + 0 * TGN_D + c4] = s_nf;
    *(float4*)&S[m * TGN_TD + 1 * TGN_D + c4] = s_te;
    *(float4*)&S[m * TGN_TD + 2 * TGN_D + c4] = s_ef;
  }

  // ---------------- Phase B: h = relu(S @ W1 + K*b1)  (fp32 WMMA) ----------------
  const int jn = lane & 15;   // output column within a 16-wide N-tile
  const int kh = lane >> 4;   // k-half selector for 16x16x4 A/B fragments

  v8f acc[8];
#pragma unroll
  for (int t = 0; t < 8; ++t)
    acc[t] = splat8(b1[t * 16 + jn] * (float)TGN_K);

  for (int ks = 0; ks < TGN_TD / 4; ++ks) {
    // A fragment: lane holds S[row=jn, k0+2*kh .. +1]  (8B-aligned -> ds_load_b64)
    const v2f a = *(const v2f*)&S[jn * TGN_TD + ks * 4 + 2 * kh];
    const float* wp0 = W1 + (ks * 4 + 2 * kh) * TGN_D + jn;
#pragma unroll
    for (int t = 0; t < 8; ++t) {
      v2f b;
      b.x = wp0[t * 16];
      b.y = wp0[t * 16 + TGN_D];
      acc[t] = __builtin_amdgcn_wmma_f32_16x16x4_f32(
          false, a, false, b, (short)0, acc[t], false, false);
    }
  }

  // ---------------- Rebuild tile in place: [relu | node_feat | cos(time_b)] ------
#pragma unroll
  for (int t = 0; t < 8; ++t)
#pragma unroll
    for (int v = 0; v < 8; ++v) {
      const int M = v + 8 * kh;                      // C/D layout row
      const float r = acc[t][v];
      S[M * TGN_TD + t * 16 + jn] = r > 0.f ? r : 0.f;
    }

  float4 ctb;
  ctb.x = fast_cos(b4.x); ctb.y = fast_cos(b4.y);
  ctb.z = fast_cos(b4.z); ctb.w = fast_cos(b4.w);
  for (int m = 0; m < 16; ++m) {
    int n = n0 + m;
    if (n >= TGN_N) n = TGN_N - 1;
    const float4 x = *(const float4*)(node_feat + (size_t)n * TGN_D + c4);
    *(float4*)&S[m * TGN_TD + 1 * TGN_D + c4] = x;     // own node features
    *(float4*)&S[m * TGN_TD + 2 * TGN_D + c4] = ctb;   // cos(time_b), row-constant
  }

  // ---------------- Phase C: out = A2 @ W2 + b2  (fp32 WMMA) ---------------------
  v8f acc2[8];
#pragma unroll
  for (int t = 0; t < 8; ++t)
    acc2[t] = splat8(b2[t * 16 + jn]);

  for (int ks = 0; ks < TGN_TD / 4; ++ks) {
    const v2f a = *(const v2f*)&S[jn * TGN_TD + ks * 4 + 2 * kh];
    const float* wp0 = W2 + (ks * 4 + 2 * kh) * TGN_D + jn;
#pragma unroll
    for (int t = 0; t < 8; ++t) {
      v2f b;
      b.x = wp0[t * 16];
      b.y = wp0[t * 16 + TGN_D];
      acc2[t] = __builtin_amdgcn_wmma_f32_16x16x4_f32(
          false, a, false, b, (short)0, acc2[t], false, false);
    }
  }

  // Guarded stores (only tail block diverges, after all WMMA are done).
#pragma unroll
  for (int t = 0; t < 8; ++t)
#pragma unroll
    for (int v = 0; v < 8; ++v) {
      const int row = n0 + v + 8 * kh;
      if (row < TGN_N)
        out[(size_t)row * TGN_D + t * 16 + jn] = acc2[t][v];
    }
}

extern "C" void kernel_launch(void* const* d_in, const int* in_sizes, int n_in,
                              void* d_out, int out_size, void* d_ws, size_t ws_size,
                              hipStream_t stream) {
  (void)in_sizes; (void)n_in; (void)out_size; (void)d_ws; (void)ws_size;
  const float* node_feat  = (const float*)d_in[0];
  const float* timestamps = (const float*)d_in[1];
  const float* edge_feat  = (const float*)d_in[2];
  const int*   neighbors  = (const int*)d_in[3];
  const int*   edge_idxs  = (const int*)d_in[4];
  const float* edge_times = (const float*)d_in[5];
  const float* time_w     = (const float*)d_in[6];
  const float* time_b     = (const float*)d_in[7];
  const float* W1         = (const float*)d_in[8];
  const float* b1         = (const float*)d_in[9];
  const float* W2         = (const float*)d_in[10];
  const float* b2         = (const float*)d_in[11];
  float* out = (float*)d_out;

  const int grid = (TGN_N + ROWS_PER_BLOCK - 1) / ROWS_PER_BLOCK;
  tgn_graphsum_fused<<<grid, WAVES_PER_BLOCK * 32, 0, stream>>>(
      node_feat, timestamps, edge_feat, neighbors, edge_idxs, edge_times,
      time_w, time_b, W1, b1, W2, b2, out);
}